// QuantumLateralCoupling_61701500174808
// MI455X (gfx1250) — compile-verified
//
#include <hip/hip_runtime.h>
#include <hip/hip_bf16.h>

#define QP  13          // qubits ("wires")
#define QD  64          // feature dim
#define NST (1 << QP)   // 8192 amplitudes
#define BLK 256         // 8 wave32 per block

typedef __attribute__((ext_vector_type(2))) float v2f;
typedef __attribute__((ext_vector_type(8))) float v8f;

// --------------------------------------------------------------------------
// Kernel 1: encoder GEMM via full-fp32 WMMA (V_WMMA_F32_16X16X4_F32).
//   angles[row, a] = sum_d h[row, d] * enc_w[a, d] + enc_b[a]
//   rows = N*QP (3328 = 208*16), K = 64 as 16 fully-unrolled k=4 WMMA steps,
//   N_out = 3 padded to 16. One wave32 per 16-row tile.
// Branch-free: addresses are clamped in-bounds (min), padding lanes are
// zeroed with a value select, so EXEC stays all-1s with no saveexec blocks.
// --------------------------------------------------------------------------
__global__ void __launch_bounds__(32)
enc_wmma_kernel(const float* __restrict__ h, const float* __restrict__ enc_w,
                const float* __restrict__ enc_b, float* __restrict__ angles,
                int rows) {
  const int lane = threadIdx.x;
  const int half = lane >> 4;     // 0: lanes 0-15, 1: lanes 16-31
  const int lm   = lane & 15;
  const int rowbase = blockIdx.x * 16;

  // ---- preload all 16 B fragments (4x16 f32 each): element (k,n) =
  //      enc_w[n*QD + k] for n<3 else 0. Clamp col, mask value. ----
  const float bmask = (lm < 3) ? 1.f : 0.f;            // v_cndmask, no branch
  const int   bcol  = (lm < 3) ? lm : 0;               // clamped address
  v2f bf[16];
#pragma unroll
  for (int kk = 0; kk < 16; ++kk) {
    // lanes 0-15: K = 4kk..4kk+1 in v0/v1; lanes 16-31: K = 4kk+2..4kk+3
    v2f bv = *(const v2f*)(enc_w + bcol * QD + kk * 4 + 2 * half);
    bf[kk].x = bv.x * bmask;
    bf[kk].y = bv.y * bmask;
  }

  // ---- A row pointer (clamped; duplicated tail rows are never stored) ----
  int arow = rowbase + lm;
  arow = (arow < rows) ? arow : (rows - 1);
  const float* aptr = h + arow * QD + 2 * half;

  v8f acc = {};
#pragma unroll
  for (int kk = 0; kk < 16; ++kk) {
    v2f a = *(const v2f*)(aptr + kk * 4);              // one b64 load
    // 8-arg f32 WMMA: (neg_a, A, neg_b, B, c_mod, C, reuse_a, reuse_b)
    acc = __builtin_amdgcn_wmma_f32_16x16x4_f32(false, a, false, bf[kk],
                                                (short)0, acc, false, false);
  }

  // D layout: VGPR j -> M = j + 8*half, N = lm (16x16 f32 C/D table)
  if (lm < 3) {
    const float bias = enc_b[lm];
#pragma unroll
    for (int j = 0; j < 8; ++j) {
      const int row = rowbase + j + (half ? 8 : 0);
      if (row < rows) angles[row * 3 + lm] = acc[j] + bias;
    }
  }
}

// --------------------------------------------------------------------------
// Kernel 2: per-sample 13-qubit state-vector simulation entirely in LDS.
// 64 KB state/block in the 320 KB WGP LDS; HBM touched once in, once out.
// --------------------------------------------------------------------------
__device__ __forceinline__ void gate2x2(float* sre, float* sim_, int bp,
                                        float u00r, float u00i,
                                        float u01r, float u01i,
                                        float u10r, float u10i,
                                        float u11r, float u11i, int t) {
  const int s = 1 << bp;
#pragma unroll 1
  for (int r = 0; r < NST / 2 / BLK; ++r) {       // 16 pairs per thread
    const int j  = t + r * BLK;                    // 0..4095
    const int i0 = ((j >> bp) << (bp + 1)) | (j & (s - 1));
    const int i1 = i0 | s;
    const float ar = sre[i0], ai = sim_[i0];
    const float br = sre[i1], bi = sim_[i1];
    sre[i0]  = u00r * ar - u00i * ai + u01r * br - u01i * bi;
    sim_[i0] = u00r * ai + u00i * ar + u01r * bi + u01i * br;
    sre[i1]  = u10r * ar - u10i * ai + u11r * br - u11i * bi;
    sim_[i1] = u10r * ai + u10i * ar + u11r * bi + u11i * br;
  }
  __syncthreads();
}

__global__ void __launch_bounds__(BLK)
qsim_kernel(const float* __restrict__ h, const float* __restrict__ angles,
            const float* __restrict__ qw, const float* __restrict__ dec_w,
            const float* __restrict__ alpha, float* __restrict__ out,
            int nlayers) {
  extern __shared__ float smem[];
  float* sre  = smem;            // 8192 re
  float* sim_ = smem + NST;      // 8192 im
  float* sang = smem + 2 * NST;  // 39 encoding angles
  float* zacc = sang + QP * 3;   // 13 <Z> accumulators

  const int n = blockIdx.x;      // sample
  const int t = threadIdx.x;
  const float* hs = h + n * (QP * QD);

  __builtin_prefetch(hs, 0, 1);  // global_prefetch_b8: warm this sample's h

  if (t < QP * 3) sang[t] = angles[n * QP * 3 + t];
  if (t < QP)     zacc[t] = 0.f;
#pragma unroll 1
  for (int k = t; k < NST; k += BLK) {             // |0...0>
    sre[k]  = (k == 0) ? 1.f : 0.f;
    sim_[k] = 0.f;
  }
  __syncthreads();

  // ---- encoding rotations: RX, RY, RZ per wire (per-sample angles) ----
#pragma unroll 1
  for (int i = 0; i < QP; ++i) {
    const int bp = QP - 1 - i;                     // wire 0 = MSB
    float th, c, s;
    th = 0.5f * sang[i * 3 + 0]; c = __cosf(th); s = __sinf(th);
    gate2x2(sre, sim_, bp, c, 0.f, 0.f, -s, 0.f, -s, c, 0.f, t);   // RX
    th = 0.5f * sang[i * 3 + 1]; c = __cosf(th); s = __sinf(th);
    gate2x2(sre, sim_, bp, c, 0.f, -s, 0.f, s, 0.f, c, 0.f, t);    // RY
    th = 0.5f * sang[i * 3 + 2]; c = __cosf(th); s = __sinf(th);
    gate2x2(sre, sim_, bp, c, -s, 0.f, 0.f, 0.f, 0.f, c, s, t);    // RZ
  }

  // ---- variational layers: RY RZ RY per wire, then ring CNOTs ----
#pragma unroll 1
  for (int L = 0; L < nlayers; ++L) {
#pragma unroll 1
    for (int i = 0; i < QP; ++i) {
      const int bp = QP - 1 - i;
      const float* w = qw + (L * QP + i) * 3;      // uniform -> scalar loads
      float th, c, s;
      th = 0.5f * w[0]; c = __cosf(th); s = __sinf(th);
      gate2x2(sre, sim_, bp, c, 0.f, -s, 0.f, s, 0.f, c, 0.f, t);  // RY
      th = 0.5f * w[1]; c = __cosf(th); s = __sinf(th);
      gate2x2(sre, sim_, bp, c, -s, 0.f, 0.f, 0.f, 0.f, c, s, t);  // RZ
      th = 0.5f * w[2]; c = __cosf(th); s = __sinf(th);
      gate2x2(sre, sim_, bp, c, 0.f, -s, 0.f, s, 0.f, c, 0.f, t);  // RY
    }
#pragma unroll 1
    for (int i = 0; i < QP; ++i) {                 // CNOT(i -> (i+1)%P)
      const int bc = QP - 1 - i;
      const int bt = QP - 1 - ((i + 1) % QP);
#pragma unroll 1
      for (int r = 0; r < NST / BLK; ++r) {
        const int idx = t + r * BLK;
        if (((idx >> bc) & 1) && !((idx >> bt) & 1)) {  // owner: ctrl=1, tgt=0
          const int ip = idx | (1 << bt);
          const float tr = sre[idx], ti = sim_[idx];
          sre[idx] = sre[ip]; sim_[idx] = sim_[ip];
          sre[ip] = tr;       sim_[ip] = ti;
        }
      }
      __syncthreads();
    }
  }

  // ---- <Z_w> = sum_i (re^2+im^2) * (bit ? -1 : +1) ----
  float z[QP];
#pragma unroll
  for (int w = 0; w < QP; ++w) z[w] = 0.f;
#pragma unroll 1
  for (int r = 0; r < NST / BLK; ++r) {
    const int idx = t + r * BLK;
    const float pr = sre[idx] * sre[idx] + sim_[idx] * sim_[idx];
#pragma unroll
    for (int w = 0; w < QP; ++w)
      z[w] += ((idx >> (QP - 1 - w)) & 1) ? -pr : pr;
  }
#pragma unroll
  for (int off = 16; off > 0; off >>= 1) {         // wave32 butterfly
#pragma unroll
    for (int w = 0; w < QP; ++w) z[w] += __shfl_xor(z[w], off, 32);
  }
  if ((t & 31) == 0) {
#pragma unroll
    for (int w = 0; w < QP; ++w) atomicAdd(&zacc[w], z[w]);  // ds_add_f32
  }
  __syncthreads();

  // ---- decode + residual: out = h + alpha * z[p] * dec_w[d] ----
  const float al = alpha[0];
#pragma unroll 1
  for (int k = t; k < QP * QD; k += BLK)
    out[n * QP * QD + k] = hs[k] + al * zacc[k >> 6] * dec_w[k & 63];
}

// --------------------------------------------------------------------------
extern "C" void kernel_launch(void* const* d_in, const int* in_sizes, int n_in,
                              void* d_out, int out_size, void* d_ws,
                              size_t ws_size, hipStream_t stream) {
  const float* h     = (const float*)d_in[0];   // (B,T,13,64) f32
  const float* enc_w = (const float*)d_in[1];   // (3,64)
  const float* enc_b = (const float*)d_in[2];   // (3,)
  const float* qw    = (const float*)d_in[3];   // (L,13,3)
  const float* dec_w = (const float*)d_in[4];   // (64,1)
  const float* alpha = (const float*)d_in[5];   // (1,)
  float* out = (float*)d_out;

  const int N       = in_sizes[0] / (QP * QD);  // B*T = 256 samples
  const int rows    = N * QP;                   // 3328 angle rows
  const int nlayers = in_sizes[3] / (QP * 3);   // 2

  float* angles = (float*)d_ws;                 // rows*3 f32 scratch (~40 KB)

  enc_wmma_kernel<<<(rows + 15) / 16, 32, 0, stream>>>(h, enc_w, enc_b,
                                                       angles, rows);

  const size_t shmem = (size_t)(2 * NST + QP * 3 + QP) * sizeof(float);
  qsim_kernel<<<N, BLK, shmem, stream>>>(h, angles, qw, dec_w, alpha, out,
                                         nlayers);
}